// GCN_Embedder_49160195670370
// MI455X (gfx1250) — compile-verified
//
#include <hip/hip_runtime.h>

// ---------------- problem constants (from reference) ----------------
constexpr int B_SZ   = 16;
constexpr int NUM_PM = 512;
constexpr int NUM_VM = 1536;
constexpr int NPG    = NUM_PM + NUM_VM;   // 2048
constexpr int N_NODE = B_SZ * NPG;        // 32768
constexpr int DEG    = 32;
constexpr int E_CNT  = DEG * N_NODE;      // 1,048,576
constexpr int D_IN   = 16;
constexpr int DH     = 64;
constexpr float NEG_SLOPE = 0.2f;

typedef float v2f __attribute__((ext_vector_type(2)));
typedef float v8f __attribute__((ext_vector_type(8)));

// order-preserving float<->uint encoding for atomicMax on floats
__device__ __forceinline__ unsigned enc_f(float f) {
    unsigned u = __float_as_uint(f);
    return (u & 0x80000000u) ? ~u : (u | 0x80000000u);
}
__device__ __forceinline__ float dec_f(unsigned u) {
    return __uint_as_float((u & 0x80000000u) ? (u & 0x7fffffffu) : ~u);
}

// ---------------- dual GEMM with f32 WMMA (D = X*W + b, twice) ----------------
// X: [n, K] row-major, W: [K, COLS] row-major, F: [n, COLS].
// One wave per 16-row strip; V_WMMA_F32_16X16X4_F32, K accumulated in steps of 4.
template <int K, int COLS>
__global__ __launch_bounds__(128) void gemm_dual_wmma(
    const float* __restrict__ X,
    const float* __restrict__ Ws, const float* __restrict__ Bs,
    const float* __restrict__ Wd, const float* __restrict__ Bd,
    float* __restrict__ Fs, float* __restrict__ Fd) {
    const int lane = threadIdx.x & 31;
    const int wave = threadIdx.x >> 5;
    const int row0 = (blockIdx.x * 4 + wave) * 16;
    const int m  = lane & 15;   // A: row M, B/D: col N
    const int hi = lane >> 4;   // lane-half selector
    const float* Xr = X + (size_t)(row0 + m) * K;

    for (int t = 0; t < 2; ++t) {
        const float* W  = t ? Wd : Ws;
        const float* Bv = t ? Bd : Bs;
        float*       F  = t ? Fd : Fs;
        for (int col0 = 0; col0 < COLS; col0 += 16) {
            v8f acc = {};
#pragma unroll
            for (int kk = 0; kk < K; kk += 4) {
                // A 16x4: lane holds M=m, K = kk + 2*hi + v
                v2f a, b;
                a.x = Xr[kk + 2 * hi + 0];
                a.y = Xr[kk + 2 * hi + 1];
                // B 4x16: lane holds N=col0+m, K = kk + 2*hi + v
                b.x = W[(size_t)(kk + 2 * hi + 0) * COLS + col0 + m];
                b.y = W[(size_t)(kk + 2 * hi + 1) * COLS + col0 + m];
                acc = __builtin_amdgcn_wmma_f32_16x16x4_f32(
                    false, a, false, b, (short)0, acc, false, false);
            }
            const float bias = Bv[col0 + m];
#pragma unroll
            for (int r = 0; r < 8; ++r) {
                // D: lane holds N=col0+m, M = r + 8*hi
                F[(size_t)(row0 + r + 8 * hi) * COLS + col0 + m] = acc[r] + bias;
            }
        }
    }
}

// ---------------- init kernels ----------------
__global__ __launch_bounds__(256) void zero_l1(float* __restrict__ h1,
                                               unsigned* __restrict__ m1,
                                               float* __restrict__ den1) {
    int idx = blockIdx.x * 256 + threadIdx.x;           // grid covers N*128
    h1[idx] = 0.f;
    if (idx < N_NODE * 2) { m1[idx] = 0u; den1[idx] = 0.f; }
}
__global__ __launch_bounds__(256) void zero_l2(float* __restrict__ agg,
                                               unsigned* __restrict__ m2,
                                               float* __restrict__ den2) {
    int idx = blockIdx.x * 256 + threadIdx.x;           // grid covers N*64
    agg[idx] = 0.f;
    if (idx < N_NODE) { m2[idx] = 0u; den2[idx] = 0.f; }
}

// ---------------- layer 1 edge kernels (2 heads x 64 feats) ----------------
// wave per edge; lane handles 4 contiguous features; lanes 0-15 = head0, 16-31 = head1
__global__ __launch_bounds__(256) void edge_logits1(
    const float* __restrict__ fs, const float* __restrict__ fd,
    const int* __restrict__ src, const int* __restrict__ dst,
    const float* __restrict__ attn, float* __restrict__ logits,
    unsigned* __restrict__ mmax) {
    const int e = blockIdx.x * 8 + (threadIdx.x >> 5);
    const int lane = threadIdx.x & 31;
    const int s = src[e], d = dst[e];
    const float4 a = *(const float4*)(fs + (size_t)s * 128 + lane * 4);
    const float4 b = *(const float4*)(fd + (size_t)d * 128 + lane * 4);
    const float4 w = *(const float4*)(attn + lane * 4);   // attn flat [2*64] matches
    float p = 0.f, t;
    t = a.x + b.x; t = t > 0.f ? t : NEG_SLOPE * t; p += t * w.x;
    t = a.y + b.y; t = t > 0.f ? t : NEG_SLOPE * t; p += t * w.y;
    t = a.z + b.z; t = t > 0.f ? t : NEG_SLOPE * t; p += t * w.z;
    t = a.w + b.w; t = t > 0.f ? t : NEG_SLOPE * t; p += t * w.w;
    p += __shfl_xor(p, 1); p += __shfl_xor(p, 2);
    p += __shfl_xor(p, 4); p += __shfl_xor(p, 8);         // reduce within 16-lane head
    if ((lane & 15) == 0) {
        const int h = lane >> 4;
        logits[(size_t)e * 2 + h] = p;
        atomicMax(&mmax[d * 2 + h], enc_f(p));
    }
}

__global__ __launch_bounds__(256) void edge_expsum1(
    float* __restrict__ logits, const int* __restrict__ dst,
    const unsigned* __restrict__ mmax, float* __restrict__ den) {
    const int idx = blockIdx.x * 256 + threadIdx.x;       // grid covers E*2
    const int e = idx >> 1, h = idx & 1;
    const int d = dst[e];
    const float ex = __expf(logits[idx] - dec_f(mmax[d * 2 + h]));
    logits[idx] = ex;                                     // store exp in place
    atomicAdd(&den[d * 2 + h], ex);
}

__global__ __launch_bounds__(256) void edge_agg1(
    const float* __restrict__ fs, const int* __restrict__ src,
    const int* __restrict__ dst, const float* __restrict__ ex,
    const float* __restrict__ den, float* __restrict__ out) {
    const int e = blockIdx.x * 8 + (threadIdx.x >> 5);
    const int lane = threadIdx.x & 31;
    const int s = src[e], d = dst[e];
    const int h = lane >> 4;
    const float alpha = ex[(size_t)e * 2 + h] / den[d * 2 + h];
    const float4 v = *(const float4*)(fs + (size_t)s * 128 + lane * 4);
    float* o = out + (size_t)d * 128 + lane * 4;
    atomicAdd(o + 0, v.x * alpha);
    atomicAdd(o + 1, v.y * alpha);
    atomicAdd(o + 2, v.z * alpha);
    atomicAdd(o + 3, v.w * alpha);
}

__global__ __launch_bounds__(256) void elu_k(float* __restrict__ h) {
    const int idx = blockIdx.x * 256 + threadIdx.x;       // grid covers N*128
    const float x = h[idx];
    h[idx] = x > 0.f ? x : (__expf(x) - 1.f);
}

// ---------------- layer 2 edge kernels (1 head x 64 feats) ----------------
__global__ __launch_bounds__(256) void edge_logits2(
    const float* __restrict__ fs, const float* __restrict__ fd,
    const int* __restrict__ src, const int* __restrict__ dst,
    const float* __restrict__ attn, float* __restrict__ logits,
    unsigned* __restrict__ mmax) {
    const int e = blockIdx.x * 8 + (threadIdx.x >> 5);
    const int lane = threadIdx.x & 31;
    const int s = src[e], d = dst[e];
    const float2 a = *(const float2*)(fs + (size_t)s * 64 + lane * 2);
    const float2 b = *(const float2*)(fd + (size_t)d * 64 + lane * 2);
    const float2 w = *(const float2*)(attn + lane * 2);
    float p = 0.f, t;
    t = a.x + b.x; t = t > 0.f ? t : NEG_SLOPE * t; p += t * w.x;
    t = a.y + b.y; t = t > 0.f ? t : NEG_SLOPE * t; p += t * w.y;
    p += __shfl_xor(p, 1);  p += __shfl_xor(p, 2);
    p += __shfl_xor(p, 4);  p += __shfl_xor(p, 8);
    p += __shfl_xor(p, 16);                               // full-wave reduce
    if (lane == 0) {
        logits[e] = p;
        atomicMax(&mmax[d], enc_f(p));
    }
}

__global__ __launch_bounds__(256) void edge_expsum2(
    float* __restrict__ logits, const int* __restrict__ dst,
    const unsigned* __restrict__ mmax, float* __restrict__ den) {
    const int e = blockIdx.x * 256 + threadIdx.x;         // grid covers E
    const int d = dst[e];
    const float ex = __expf(logits[e] - dec_f(mmax[d]));
    logits[e] = ex;
    atomicAdd(&den[d], ex);
}

__global__ __launch_bounds__(256) void edge_agg2(
    const float* __restrict__ fs, const int* __restrict__ src,
    const int* __restrict__ dst, const float* __restrict__ ex,
    const float* __restrict__ den, float* __restrict__ out) {
    const int e = blockIdx.x * 8 + (threadIdx.x >> 5);
    const int lane = threadIdx.x & 31;
    const int s = src[e], d = dst[e];
    const float alpha = ex[e] / den[d];
    const float2 v = *(const float2*)(fs + (size_t)s * 64 + lane * 2);
    float* o = out + (size_t)d * 64 + lane * 2;
    atomicAdd(o + 0, v.x * alpha);
    atomicAdd(o + 1, v.y * alpha);
}

// ---------------- final reorder into (PM block, VM block) ----------------
__global__ __launch_bounds__(256) void reorder_out(const float* __restrict__ agg,
                                                   float* __restrict__ out) {
    const int idx = blockIdx.x * 256 + threadIdx.x;       // grid covers N*64
    const int n = idx >> 6, f = idx & 63;
    const int b = n >> 11;           // / NPG
    const int r = n & 2047;          // % NPG
    size_t o;
    if (r < NUM_PM) o = ((size_t)(b * NUM_PM + r) * 64) + f;
    else            o = (size_t)B_SZ * NUM_PM * 64 +
                        ((size_t)(b * NUM_VM + (r - NUM_PM)) * 64) + f;
    out[o] = agg[idx];
}

// ---------------- host launcher ----------------
extern "C" void kernel_launch(void* const* d_in, const int* in_sizes, int n_in,
                              void* d_out, int out_size, void* d_ws, size_t ws_size,
                              hipStream_t stream) {
    const float* x     = (const float*)d_in[0];
    const int*   src   = (const int*)d_in[1];
    const int*   dst   = (const int*)d_in[2];
    const float* W1s   = (const float*)d_in[3];
    const float* b1s   = (const float*)d_in[4];
    const float* W1d   = (const float*)d_in[5];
    const float* b1d   = (const float*)d_in[6];
    const float* attn1 = (const float*)d_in[7];
    const float* W2s   = (const float*)d_in[8];
    const float* b2s   = (const float*)d_in[9];
    const float* W2d   = (const float*)d_in[10];
    const float* b2d   = (const float*)d_in[11];
    const float* attn2 = (const float*)d_in[12];
    (void)in_sizes; (void)n_in; (void)out_size; (void)ws_size;

    char* ws = (char*)d_ws;
    const size_t MB = 1024u * 1024u;
    // phase-1 layout (peak 57 MB)
    float*    fs1   = (float*)(ws + 0);                 // 16 MB [N,128]
    float*    fd1   = (float*)(ws + 16 * MB);           // 16 MB [N,128]
    float*    log1  = (float*)(ws + 32 * MB);           //  8 MB [E,2]
    unsigned* m1    = (unsigned*)(ws + 40 * MB);        // 256 KB [N,2]
    float*    den1  = (float*)(ws + 40 * MB + 262144);  // 256 KB [N,2]
    float*    h1    = (float*)(ws + 41 * MB);           // 16 MB [N,128]
    // phase-2 layout (reuses fs1/fd1 regions, dead by then)
    float*    fs2   = (float*)(ws + 16 * MB);           //  8 MB [N,64]
    float*    fd2   = (float*)(ws + 24 * MB);           //  8 MB [N,64]
    float*    agg2  = (float*)(ws + 0);                 //  8 MB [N,64]
    float*    log2  = (float*)(ws + 8 * MB);            //  4 MB [E]
    unsigned* m2    = (unsigned*)(ws + 12 * MB);        // 128 KB [N]
    float*    den2  = (float*)(ws + 12 * MB + 131072);  // 128 KB [N]

    float* out = (float*)d_out;

    // ---- layer 1 ----
    zero_l1<<<N_NODE * 128 / 256, 256, 0, stream>>>(h1, m1, den1);
    gemm_dual_wmma<D_IN, 2 * DH><<<N_NODE / 16 / 4, 128, 0, stream>>>(
        x, W1s, b1s, W1d, b1d, fs1, fd1);
    edge_logits1<<<E_CNT / 8, 256, 0, stream>>>(fs1, fd1, src, dst, attn1, log1, m1);
    edge_expsum1<<<E_CNT * 2 / 256, 256, 0, stream>>>(log1, dst, m1, den1);
    edge_agg1<<<E_CNT / 8, 256, 0, stream>>>(fs1, src, dst, log1, den1, h1);
    elu_k<<<N_NODE * 128 / 256, 256, 0, stream>>>(h1);

    // ---- layer 2 ----
    zero_l2<<<N_NODE * 64 / 256, 256, 0, stream>>>(agg2, m2, den2);
    gemm_dual_wmma<2 * DH, DH><<<N_NODE / 16 / 4, 128, 0, stream>>>(
        h1, W2s, b2s, W2d, b2d, fs2, fd2);
    edge_logits2<<<E_CNT / 8, 256, 0, stream>>>(fs2, fd2, src, dst, attn2, log2, m2);
    edge_expsum2<<<E_CNT / 256, 256, 0, stream>>>(log2, dst, m2, den2);
    edge_agg2<<<E_CNT / 8, 256, 0, stream>>>(fs2, src, dst, log2, den2, agg2);

    // ---- split into (PM, VM) output blocks ----
    reorder_out<<<N_NODE * 64 / 256, 256, 0, stream>>>(agg2, out);
}